// SHRender_42872363549157
// MI455X (gfx1250) — compile-verified
//
#include <hip/hip_runtime.h>
#include <cmath>

typedef float v2f __attribute__((ext_vector_type(2)));
typedef float v8f __attribute__((ext_vector_type(8)));

#define EPSV   1e-10f
#define HREZ   128
#define WREZ   128
#define RECSZ  24          // floats per face record (96 B, 16B-aligned)
#define CF     112         // faces per LDS chunk (multiple of 16)

// -------- face record layout (floats) --------
// WMMA B block, lane-addressed (lane base = rec + (hi ? 6 : 0)):
//  0: c0a  1: c0x   | 6: c0y  7: 0     -> B0 = load64
//  2: c1a  3: c1x   | 8: c1y  9: 0     -> B1 = load64
//  4: za   5: zx    |10: zy  11: 0     -> BZ = load64
// 12   : ok flag (valid && |area|>eps)
// 13..15: raw face normal (nx,ny,nz)
// 16..21: u0,v0,u1,v1,u2,v2
// 22..23: pad

__global__ void __launch_bounds__(256)
face_setup_kernel(const float* __restrict__ pts, const int* __restrict__ faces,
                  const float* __restrict__ rot, const float* __restrict__ cpos,
                  const float* __restrict__ proj, const float* __restrict__ uv,
                  const int* __restrict__ ft, float* __restrict__ fd,
                  float* __restrict__ nrm_out, int F, int FPAD)
{
  int f = blockIdx.x * blockDim.x + threadIdx.x;
  if (f >= FPAD) return;
  float* rec = fd + f * RECSZ;
  if (f >= F) {
    #pragma unroll
    for (int i = 0; i < RECSZ; ++i) rec[i] = 0.f;
    return;
  }
  const int i0 = faces[f * 3 + 0], i1 = faces[f * 3 + 1], i2 = faces[f * 3 + 2];
  const float r00 = rot[0], r01 = rot[1], r02 = rot[2];
  const float r10 = rot[3], r11 = rot[4], r12 = rot[5];
  const float r20 = rot[6], r21 = rot[7], r22 = rot[8];
  const float cx = cpos[0], cy = cpos[1], cz = cpos[2];

  float p0[3], p1[3], p2[3];
  auto camv = [&](int idx, float* o) {
    float ax = pts[idx * 3 + 0] - cx;
    float ay = pts[idx * 3 + 1] - cy;
    float az = pts[idx * 3 + 2] - cz;
    o[0] = r00 * ax + r01 * ay + r02 * az;
    o[1] = r10 * ax + r11 * ay + r12 * az;
    o[2] = r20 * ax + r21 * ay + r22 * az;
  };
  camv(i0, p0); camv(i1, p1); camv(i2, p2);

  const float pj0 = proj[0], pj1 = proj[1], pj2 = proj[2];
  const float x0 = (p0[0] * pj0) / (p0[2] * pj2), y0 = (p0[1] * pj1) / (p0[2] * pj2);
  const float x1 = (p1[0] * pj0) / (p1[2] * pj2), y1 = (p1[1] * pj1) / (p1[2] * pj2);
  const float x2 = (p2[0] * pj0) / (p2[2] * pj2), y2 = (p2[1] * pj1) / (p2[2] * pj2);

  // face normal in camera space
  const float e1x = p1[0] - p0[0], e1y = p1[1] - p0[1], e1z = p1[2] - p0[2];
  const float e2x = p2[0] - p0[0], e2y = p2[1] - p0[1], e2z = p2[2] - p0[2];
  const float nx = e1y * e2z - e1z * e2y;
  const float ny = e1z * e2x - e1x * e2z;
  const float nz = e1x * e2y - e1y * e2x;
  const float nl = sqrtf(nx * nx + ny * ny + nz * nz);
  const float ninv = 1.f / fmaxf(nl, EPSV);
  nrm_out[f * 3 + 0] = nx * ninv;
  nrm_out[f * 3 + 1] = ny * ninv;
  nrm_out[f * 3 + 2] = nz * ninv;
  const bool valid = nz > 0.f;

  // edge functions
  const float A0 = x1 * y2 - x2 * y1, Bx0 = y1 - y2, By0 = x2 - x1;
  const float A1 = x2 * y0 - x0 * y2, Bx1 = y2 - y0, By1 = x0 - x2;
  const float A2 = x0 * y1 - x1 * y0;
  const float area = A0 + A1 + A2;
  const bool areaok = fabsf(area) > EPSV;
  const float ia = areaok ? 1.f / area : 0.f;
  const bool ok = valid && areaok;

  const float c0a = A0 * ia, c0x = Bx0 * ia, c0y = By0 * ia;
  const float c1a = A1 * ia, c1x = Bx1 * ia, c1y = By1 * ia;
  const float dz0 = p0[2] - p2[2], dz1 = p1[2] - p2[2];

  // lo-lane block
  rec[0] = c0a; rec[1] = c0x;
  rec[2] = c1a; rec[3] = c1x;
  rec[4] = p2[2] + c0a * dz0 + c1a * dz1;
  rec[5] = c0x * dz0 + c1x * dz1;
  // hi-lane block (zeros in the K=3 slots)
  rec[6] = c0y; rec[7] = 0.f;
  rec[8] = c1y; rec[9] = 0.f;
  rec[10] = c0y * dz0 + c1y * dz1; rec[11] = 0.f;

  rec[12] = ok ? 1.f : 0.f;
  rec[13] = nx; rec[14] = ny; rec[15] = nz;
  const int t0 = ft[f * 3 + 0], t1 = ft[f * 3 + 1], t2 = ft[f * 3 + 2];
  rec[16] = uv[t0 * 2 + 0]; rec[17] = uv[t0 * 2 + 1];
  rec[18] = uv[t1 * 2 + 0]; rec[19] = uv[t1 * 2 + 1];
  rec[20] = uv[t2 * 2 + 0]; rec[21] = uv[t2 * 2 + 1];
  rec[22] = 0.f; rec[23] = 0.f;
}

// One block per image row. 8 waves x 16 pixels = 128 pixels.
// Per wave: A = [16 px x 4] with columns [1, px, py, 0];
// per 16-face batch: 3x V_WMMA_F32_16X16X4_F32 -> b0, b1, zpix tiles.
__global__ void __launch_bounds__(256)
raster_kernel(const float* __restrict__ fd, const float* __restrict__ tex,
              const float* __restrict__ light, float* __restrict__ out_render,
              float* __restrict__ out_prob, int FPAD, int TH, int TW)
{
  __shared__ __align__(16) float shFace[CF * RECSZ];
  __shared__ float shS[8 * 16];
  __shared__ int   shI[8 * 16];
  __shared__ float shP[8 * 16];

  const int tid  = threadIdx.x;
  const int wave = tid >> 5;
  const int lane = tid & 31;
  const int col  = lane & 15;
  const bool hi  = lane >= 16;
  const int y    = blockIdx.x;
  const int xbase = wave * 16;

  const float pyv = 1.f - (float)y * (2.f / (HREZ - 1));
  const float pxm = -1.f + (float)(xbase + col) * (2.f / (WREZ - 1));

  // A-matrix (16x4 f32): lanes 0-15 -> K=0 (1.0) / K=1 (px); lanes 16-31 -> K=2 (py) / K=3 (0)
  v2f A;
  A.x = hi ? pyv : 1.f;
  A.y = hi ? 0.f : pxm;

  float bestS[8], prod[8];
  int   bestI[8];
  #pragma unroll
  for (int r = 0; r < 8; ++r) { bestS[r] = -1.0e9f; bestI[r] = 0; prod[r] = 1.f; }

  for (int base = 0; base < FPAD; base += CF) {
    const int cnt = min(CF, FPAD - base);
    __syncthreads();
    for (int i = tid; i < cnt * RECSZ; i += 256)
      shFace[i] = fd[base * RECSZ + i];
    if (base + CF < FPAD) {
      int pf = (base + CF) * RECSZ + tid * 12;
      if (pf < FPAD * RECSZ) __builtin_prefetch(fd + pf, 0, 1);
    }
    __syncthreads();

    const int nb = cnt >> 4;
    for (int b = 0; b < nb; ++b) {
      const float* recb = &shFace[(b * 16 + col) * RECSZ];
      const float* lrec = recb + (hi ? 6 : 0);     // lane's B block
      const int fglob = base + b * 16 + col;
      // B-matrix (4x16 f32): VGPR0 lanes0-15=K0, lanes16-31=K2; VGPR1 lanes0-15=K1, lanes16-31=K3
      const v2f B0 = *(const v2f*)(lrec + 0);
      const v2f B1 = *(const v2f*)(lrec + 2);
      const v2f BZ = *(const v2f*)(lrec + 4);
      const bool okb = recb[12] > 0.5f;

      v8f zero = {0.f, 0.f, 0.f, 0.f, 0.f, 0.f, 0.f, 0.f};
      v8f D0 = __builtin_amdgcn_wmma_f32_16x16x4_f32(false, A, false, B0, (short)0, zero, false, false);
      v8f D1 = __builtin_amdgcn_wmma_f32_16x16x4_f32(false, A, false, B1, (short)0, zero, false, false);
      v8f DZ = __builtin_amdgcn_wmma_f32_16x16x4_f32(false, A, false, BZ, (short)0, zero, false, false);

      #pragma unroll
      for (int r = 0; r < 8; ++r) {
        const float b0 = D0[r], b1 = D1[r];
        const float b2 = 1.f - b0 - b1;
        const float z  = DZ[r];
        const bool inside = okb && (b0 >= 0.f) && (b1 >= 0.f) && (b2 >= 0.f);
        const float score = inside ? z : -1.0e9f;
        if (score > bestS[r]) { bestS[r] = score; bestI[r] = fglob; }
        const float bmin = fminf(fminf(b0, b1), b2);
        const float d = fmaxf(-bmin, 0.f);
        const float pr = okb ? __expf(-d * d * 1.0e4f) : 0.f;
        prod[r] *= (1.f - pr);
      }
    }
  }

  // butterfly reduce over the 16 face-columns (groups of 16 lanes)
  #pragma unroll
  for (int r = 0; r < 8; ++r) {
    float s = bestS[r]; int fi = bestI[r]; float p = prod[r];
    #pragma unroll
    for (int mask = 1; mask < 16; mask <<= 1) {
      float os = __shfl_xor(s, mask, 16);
      int   oi = __shfl_xor(fi, mask, 16);
      float op = __shfl_xor(p, mask, 16);
      p *= op;
      if (os > s || (os == s && oi < fi)) { s = os; fi = oi; }
    }
    bestS[r] = s; bestI[r] = fi; prod[r] = p;
  }
  if (col == 0) {
    #pragma unroll
    for (int r = 0; r < 8; ++r) {
      const int row = r + (hi ? 8 : 0);
      shS[wave * 16 + row] = bestS[r];
      shI[wave * 16 + row] = bestI[r];
      shP[wave * 16 + row] = prod[r];
    }
  }
  __syncthreads();

  if (lane < 16) {
    const int x = xbase + lane;
    const float pxv = -1.f + (float)x * (2.f / (WREZ - 1));
    const float s = shS[wave * 16 + lane];
    const int  fi = shI[wave * 16 + lane];
    const float p = shP[wave * 16 + lane];
    const float visible = (s > -5.0e8f) ? 1.f : 0.f;

    const float* rec = fd + fi * RECSZ;
    const float w0 = rec[0] + pxv * rec[1] + pyv * rec[6];
    const float w1 = rec[2] + pxv * rec[3] + pyv * rec[8];
    const float w2 = 1.f - w0 - w1;

    const float nx = rec[13] * visible, ny = rec[14] * visible, nz = rec[15] * visible;
    const float u = (w0 * rec[16] + w1 * rec[18] + w2 * rec[20]) * visible;
    const float v = (w0 * rec[17] + w1 * rec[19] + w2 * rec[21]) * visible;
    const float hard = (w0 + w1 + w2) * visible;

    const float len = sqrtf(nx * nx + ny * ny + nz * nz);
    const float inv = 1.f / fmaxf(len, EPSV);
    const float X = nx * inv, Y = ny * inv, Z = nz * inv;

    const float shading =
        0.2820948f * light[0] +
        0.4886025f * Y * light[1] +
        0.4886025f * Z * light[2] +
        0.4886025f * X * light[3] +
        1.092548f  * X * Y * light[4] +
        1.092548f  * Y * Z * light[5] +
        0.3153916f * (3.f * Z * Z - 1.f) * light[6] +
        1.092548f  * X * Z * light[7] +
        0.5462742f * (X * X - Y * Y) * light[8];

    const float uc = fminf(fmaxf(u, 0.f), 1.f) * (float)(TW - 1);
    const float vc = fminf(fmaxf(v, 0.f), 1.f) * (float)(TH - 1);
    const float x0f = floorf(uc), y0f = floorf(vc);
    const int x0i = (int)x0f, y0i = (int)y0f;
    const int x1i = min(x0i + 1, TW - 1);
    const int y1i = min(y0i + 1, TH - 1);
    const float fx = uc - x0f, fy = vc - y0f;

    const int pix = y * WREZ + x;
    #pragma unroll
    for (int c = 0; c < 3; ++c) {
      const float* tc = tex + c * TH * TW;
      const float c00 = tc[y0i * TW + x0i];
      const float c01 = tc[y0i * TW + x1i];
      const float c10 = tc[y1i * TW + x0i];
      const float c11 = tc[y1i * TW + x1i];
      const float texc = (c00 * (1.f - fx) + c01 * fx) * (1.f - fy)
                       + (c10 * (1.f - fx) + c11 * fx) * fy;
      const float val = texc * shading * hard;
      out_render[pix * 3 + c] = fminf(fmaxf(val, 0.f), 1.f);
    }
    out_prob[pix] = 1.f - p;
  }
}

extern "C" void kernel_launch(void* const* d_in, const int* in_sizes, int n_in,
                              void* d_out, int out_size, void* d_ws, size_t ws_size,
                              hipStream_t stream) {
  const float* pts  = (const float*)d_in[0];
  const int*   faces= (const int*)  d_in[1];
  const float* rot  = (const float*)d_in[2];
  const float* cpos = (const float*)d_in[3];
  const float* proj = (const float*)d_in[4];
  const float* uv   = (const float*)d_in[5];
  const int*   ft   = (const int*)  d_in[6];
  const float* tex  = (const float*)d_in[7];
  const float* light= (const float*)d_in[8];
  float* out = (float*)d_out;

  const int F = in_sizes[1] / 3;
  const int FPAD = (F + 15) & ~15;
  const int thw = in_sizes[7] / 3;
  const int TH = (int)(sqrt((double)thw) + 0.5);
  const int TW = TH;

  float* fdata = (float*)d_ws;  // FPAD * RECSZ floats

  float* out_render = out;                          // H*W*3
  float* out_prob   = out + HREZ * WREZ * 3;        // H*W
  float* out_norm   = out + HREZ * WREZ * 4;        // F*3

  face_setup_kernel<<<(FPAD + 255) / 256, 256, 0, stream>>>(
      pts, faces, rot, cpos, proj, uv, ft, fdata, out_norm, F, FPAD);
  raster_kernel<<<HREZ, 256, 0, stream>>>(
      fdata, tex, light, out_render, out_prob, FPAD, TH, TW);
}